// DWT1D_257698038000
// MI455X (gfx1250) — compile-verified
//
#include <hip/hip_runtime.h>
#include <stdint.h>

#define N_IN   16384   // samples per row
#define L_OUT  8194    // (N + 6 - 1) / 2
#define C_CH   64      // input channels

// --- TDM vector types (match probe-confirmed builtin signatures) ---
typedef unsigned int  tdm_v4u __attribute__((ext_vector_type(4)));
typedef int           tdm_v8i __attribute__((ext_vector_type(8)));
typedef int           tdm_v4i __attribute__((ext_vector_type(4)));
typedef float         v2f     __attribute__((ext_vector_type(2)));

// Half-sample symmetric reflection: i<0 -> -i-1 ; i>=n -> 2n-1-i
__device__ __forceinline__ int symref(int i, int n) {
    i = (i < 0) ? (-i - 1) : i;
    i = (i >= n) ? (2 * n - 1 - i) : i;
    return i;
}

__global__ void dwt_ds_kernel(const float* __restrict__ x,
                              float* __restrict__ out) {
    extern __shared__ float s_x[];            // 16384 floats = 64 KB

    const int row = blockIdx.x;               // row = b*64 + c
    const float* __restrict__ xrow = x + (size_t)row * N_IN;

#if __has_builtin(__builtin_amdgcn_tensor_load_to_lds)
    // Uniform scalar branch: only wave 0 issues the TDM DMA (TDM ignores EXEC,
    // so the other waves must *branch around* it, not just be masked).
    unsigned wid = ((unsigned)__builtin_amdgcn_readfirstlane((int)threadIdx.x)) >> 5;
    if (wid == 0) {
        const uint64_t ga       = (uint64_t)(uintptr_t)xrow;
        const uint32_t lds_base = (uint32_t)(uintptr_t)&s_x[0]; // low 32b of flat LDS addr = LDS offset

        // ---- D# group 0 (128b): count=1 | lds_addr | global_addr[56:0] | type=2 ----
        tdm_v4u g0;
        g0.x = 1u;                                              // count=1, user descriptor
        g0.y = lds_base;                                        // lds_addr (bytes)
        g0.z = (uint32_t)(ga & 0xFFFFFFFFu);                    // global_addr[31:0]
        g0.w = (uint32_t)((ga >> 32) & 0x01FFFFFFu)             // global_addr[56:32]
             | (2u << 30);                                      // type = 2 ("image")

        // ---- D# group 1 (256b): data_size=4B, 1 x 16384 tile == tensor ----
        tdm_v8i g1;
        g1[0] = (int)(2u << 16);                 // workgroup_mask=0 | data_size=2 (4 bytes)
        g1[1] = (int)((uint32_t)N_IN << 16);     // atomic_barrier_addr=0 | tensor_dim0[15:0]
        g1[2] = (int)(1u << 16);                 // tensor_dim0[31:16]=0 | tensor_dim1[15:0]=1
        g1[3] = (int)((uint32_t)N_IN << 16);     // tensor_dim1[31:16]=0 | tile_dim0=16384
        g1[4] = 1;                               // tile_dim1=1 | tile_dim2=0
        g1[5] = N_IN;                            // tensor_dim0_stride[31:0]
        g1[6] = (int)((uint32_t)N_IN << 16);     // stride0[47:32]=0 | tensor_dim1_stride[15:0]
        g1[7] = 0;                               // tensor_dim1_stride[47:16]=0

        tdm_v4i gz = {0, 0, 0, 0};               // groups 2/3 unused (<=2D tensor)
#if __has_include(<hip/amd_detail/amd_gfx1250_TDM.h>)
        tdm_v8i gz8 = {0, 0, 0, 0, 0, 0, 0, 0};
        __builtin_amdgcn_tensor_load_to_lds(g0, g1, gz, gz, gz8, 0);   // clang-23 / therock: 6 args
#else
        __builtin_amdgcn_tensor_load_to_lds(g0, g1, gz, gz, 0);        // ROCm 7.2 clang-22: 5 args
#endif
        __builtin_amdgcn_s_wait_tensorcnt(0);
    }
    __syncthreads();
#else
    // Fallback: cooperative vectorized stage into LDS.
    {
        const float4* __restrict__ src4 = (const float4*)xrow;
        float4* dst4 = (float4*)s_x;
        for (int i = threadIdx.x; i < N_IN / 4; i += blockDim.x)
            dst4[i] = src4[i];
    }
    __syncthreads();
#endif

    // Output layout: out[b, c', k], c' in [0,192): [ds | a | d] blocks of 64.
    const int b = row / C_CH;
    const int c = row % C_CH;
    float* __restrict__ outb = out + (size_t)b * (3 * C_CH) * L_OUT;
    float* __restrict__ o_ds = outb + (size_t)(c)             * L_OUT;
    float* __restrict__ o_a  = outb + (size_t)(C_CH + c)      * L_OUT;
    float* __restrict__ o_d  = outb + (size_t)(2 * C_CH + c)  * L_OUT;

    // bior2.2 decomposition filters
    const float kLo[6] = { 0.0f, -0.1767766952966369f, 0.3535533905932738f,
                           1.0606601717798212f, 0.3535533905932738f, -0.1767766952966369f };
    const float kHi[6] = { 0.0f, 0.3535533905932738f, -0.7071067811865476f,
                           0.3535533905932738f, 0.0f, 0.0f };

    const float scale = (float)N_IN / (float)L_OUT;

    // Pair-wise: L_OUT = 8194 = 2*4097 -> no tail; element index c'*8194 + k0
    // is always even -> every v2f store is 8-byte aligned.
    for (int k0 = 2 * (int)threadIdx.x; k0 < L_OUT; k0 += 2 * (int)blockDim.x) {
        v2f ra, rd, rds;
#pragma unroll
        for (int u = 0; u < 2; ++u) {
            const int k = k0 + u;

            // ---- DWT: a[k] = sum_m kLo[m]*x[sym(2k+1-m)], d[k] likewise ----
            const int base = 2 * k + 1;
            float acc_a = 0.0f, acc_d = 0.0f;
#pragma unroll
            for (int m = 0; m < 6; ++m) {
                const float v = s_x[symref(base - m, N_IN)];
                acc_a = fmaf(kLo[m], v, acc_a);
                acc_d = fmaf(kHi[m], v, acc_d);
            }

            // ---- linear resample (align_corners=False), fp32 as in reference ----
            float src = ((float)k + 0.5f) * scale - 0.5f;
            src = fminf(fmaxf(src, 0.0f), (float)(N_IN - 1));
            const int   i0 = (int)src;                  // src >= 0 -> trunc == floor
            const int   i1 = min(i0 + 1, N_IN - 1);
            const float w  = src - (float)i0;
            const float ds = s_x[i0] * (1.0f - w) + s_x[i1] * w;

            ra[u]  = acc_a;
            rd[u]  = acc_d;
            rds[u] = ds;
        }
        // Streaming (non-temporal) B64 stores: written once, never re-read.
        __builtin_nontemporal_store(rds, (v2f*)(o_ds + k0));
        __builtin_nontemporal_store(ra,  (v2f*)(o_a  + k0));
        __builtin_nontemporal_store(rd,  (v2f*)(o_d  + k0));
    }
}

extern "C" void kernel_launch(void* const* d_in, const int* in_sizes, int n_in,
                              void* d_out, int out_size, void* d_ws, size_t ws_size,
                              hipStream_t stream) {
    (void)n_in; (void)out_size; (void)d_ws; (void)ws_size;
    const float* x = (const float*)d_in[0];
    float* out = (float*)d_out;

    const int rows = in_sizes[0] / N_IN;   // 32 * 64 = 2048
    dim3 grid((unsigned)rows), block(256);
    const size_t shmem = (size_t)N_IN * sizeof(float);  // 64 KB/row staged via TDM

    hipLaunchKernelGGL(dwt_ds_kernel, grid, block, shmem, stream, x, out);
}